// mpNet_Constrained_1486058684580
// MI455X (gfx1250) — compile-verified
//
#include <hip/hip_runtime.h>

// MI455X (gfx1250) matching-pursuit kernel.
// Problem constants fixed by the harness's setup_inputs():
//   B = 8192 samples, N = 64 antennas, A = 1024 atoms, k = 8 iterations.
#define NB 8192
#define NN 64
#define NA 1024

typedef __attribute__((ext_vector_type(2))) float v2f;
typedef __attribute__((ext_vector_type(8))) float v8f;

#define TWO_PI_F 6.28318530717958647692f
#define LAM_F    0.1f

// ---------------------------------------------------------------------------
// Kernel 1: build Wc = conj(W), packed in WMMA-B-fragment-native layout:
//   wcp[kp*1024 + a] = float4{ Re(2kp,a), Re(2kp+1,a), Im(2kp,a), Im(2kp+1,a) }
// so a lane's whole (re,im) B fragment for a K-pair is ONE b128 load whose
// sub-register pairs are directly usable as even-aligned v2f WMMA operands.
// Total 512 KB in d_ws; fully L2/WGP$-resident.
// ---------------------------------------------------------------------------
__global__ void build_wc_kernel(const float* __restrict__ antpos, // [64,3]
                                const float* __restrict__ doa,    // [1024,3]
                                const float* __restrict__ gvec,   // [64]
                                float* __restrict__ wcp)
{
    const int idx = blockIdx.x * blockDim.x + threadIdx.x;   // n*1024 + a
    const int n = idx >> 10;
    const int a = idx & (NA - 1);
    const float d = antpos[n * 3 + 0] * doa[a * 3 + 0]
                  + antpos[n * 3 + 1] * doa[a * 3 + 1]
                  + antpos[n * 3 + 2] * doa[a * 3 + 2];
    const float phase = (-TWO_PI_F / LAM_F) * d;
    float s, c;
    __sincosf(phase, &s, &c);
    const float g = gvec[n];
    const int kp = n >> 1;          // K row-pair
    const int hh = n & 1;           // position within pair
    float* base = wcp + ((long)kp * NA + a) * 4;
    base[hh]     =  g * c;          // Re(conj(W))
    base[2 + hh] = -g * s;          // Im(conj(W))
}

// ---------------------------------------------------------------------------
// Kernel 2: fused k-iteration matching pursuit.
// One wave owns 16 residual rows as WMMA A-fragments (registers) for the
// whole kernel. Per iteration: 3-mult complex GEMM via V_WMMA_F32_16X16X4_F32
//   t1 = (ar+ai)@wr, t2 = ai@(wr+wi), t3 = ar@(wi-wr)
//   cre = t1-t2, cim = t1+t3
// then per-row argmax of |c|^2 (shuffle butterfly) and 1-sparse register
// update. All 8 iterations in one launch; no inter-block dependency.
// ---------------------------------------------------------------------------
__global__ __launch_bounds__(128, 1)
void mp_fused_kernel(const float* __restrict__ x,     // interleaved c64 [B,64]
                     const float4* __restrict__ wcp,  // packed dict [32][1024]
                     const int*   __restrict__ kptr,  // scalar k (=8)
                     float* __restrict__ out_res,     // interleaved c64 [B,64]
                     float* __restrict__ out_xhat)    // interleaved c64 [B,64]
{
    const int lane = threadIdx.x & 31;
    const int wave = threadIdx.x >> 5;       // 4 waves / block
    const int h    = lane >> 4;              // half-wave: selects K pair (A/B layout)
    const int m    = lane & 15;              // A-row / B-col within 16x16 tile
    const long rowBase = ((long)blockIdx.x * 4 + wave) * 16;
    const long b = rowBase + m;              // this lane's sample row

    __shared__ int   s_idx[4][16];
    __shared__ float s_re[4][16];
    __shared__ float s_im[4][16];

    // Warm the WGP$/L2 with the packed dictionary (global_prefetch_b8),
    // one 128B line per prefetch, 512 KB total per block.
    for (int p = threadIdx.x; p < 4096; p += 128)
        __builtin_prefetch((const float*)wcp + p * 32, 0, 0);

    // Load residual rows as A-fragments (16x4 f32 layout):
    // lane holds row m; k-chunk kc covers K = 4*kc + 2*h, +1  (VGPR0/VGPR1).
    v2f a_re[16], a_im[16], a_sum[16];
#pragma unroll
    for (int kc = 0; kc < 16; ++kc) {
        const int k0 = 4 * kc + 2 * h;
        const float4 v = *(const float4*)(x + (b * NN + k0) * 2);
        a_re[kc] = (v2f){v.x, v.z};
        a_im[kc] = (v2f){v.y, v.w};
    }

    const int K = *kptr;   // uniform scalar load

    for (int it = 0; it < K; ++it) {
        // Karatsuba A-side combo (recomputed each iteration after the update).
#pragma unroll
        for (int kc = 0; kc < 16; ++kc) a_sum[kc] = a_re[kc] + a_im[kc];

        // Running per-slot best: slot e <-> row (e + 8*h), columns (16*t + m).
        float bmag[8], bre[8], bim[8];
        int   bidx[8];
#pragma unroll
        for (int e = 0; e < 8; ++e) { bmag[e] = -1.0f; bidx[e] = 0; bre[e] = 0.0f; bim[e] = 0.0f; }

        for (int t = 0; t < 64; ++t) {
            const int col = t * 16 + m;
            v8f acc1 = {}, acc2 = {}, acc3 = {};
#pragma unroll
            for (int kc = 0; kc < 16; ++kc) {
                // One b128 load per chunk: {re0, re1, im0, im1} for K pair.
                const float4 q = wcp[(2 * kc + h) * NA + col];
                const v2f b_r = (v2f){q.x, q.y};                  // wr
                const v2f b_s = (v2f){q.x + q.z, q.y + q.w};      // wr + wi
                const v2f b_d = (v2f){q.z - q.x, q.w - q.y};      // wi - wr
                acc1 = __builtin_amdgcn_wmma_f32_16x16x4_f32(false, a_sum[kc], false, b_r,
                                                             (short)0, acc1, false, false);
                acc2 = __builtin_amdgcn_wmma_f32_16x16x4_f32(false, a_im[kc], false, b_s,
                                                             (short)0, acc2, false, false);
                acc3 = __builtin_amdgcn_wmma_f32_16x16x4_f32(false, a_re[kc], false, b_d,
                                                             (short)0, acc3, false, false);
            }
            // cre = t1 - t2, cim = t1 + t3.
#pragma unroll
            for (int e = 0; e < 8; ++e) {
                const float cre = acc1[e] - acc2[e];
                const float cim = acc1[e] + acc3[e];
                const float mag = cre * cre + cim * cim;
                if (mag > bmag[e]) { bmag[e] = mag; bidx[e] = col; bre[e] = cre; bim[e] = cim; }
            }
        }

        // Per-row reduction across the 16 lanes of each half-wave.
        // First-occurrence argmax: prefer strictly larger mag, lower idx on tie.
#pragma unroll
        for (int e = 0; e < 8; ++e) {
            float mg = bmag[e]; int ix = bidx[e]; float cr = bre[e]; float ci = bim[e];
#pragma unroll
            for (int off = 1; off < 16; off <<= 1) {
                const float omg = __shfl_xor(mg, off, 16);
                const int   oix = __shfl_xor(ix, off, 16);
                const float ocr = __shfl_xor(cr, off, 16);
                const float oci = __shfl_xor(ci, off, 16);
                if (omg > mg || (omg == mg && oix < ix)) { mg = omg; ix = oix; cr = ocr; ci = oci; }
            }
            if (m == 0) {                 // one writer per half-wave
                const int row = e + 8 * h;
                s_idx[wave][row] = ix;
                s_re[wave][row]  = cr;
                s_im[wave][row]  = ci;
            }
        }
        // Intra-wave LDS RAW: compiler inserts s_wait_dscnt; no barrier needed.
        const int   selidx = s_idx[wave][m];
        const float sre    = s_re[wave][m];
        const float sim    = s_im[wave][m];

        // residual -= sel * atom, atom = W[:,idx] = conj(Wc[:,idx]):
        //   atom_re = q.re, atom_im = -q.im
        //   d_re = sre*q.re + sim*q.im ; d_im = sim*q.re - sre*q.im
#pragma unroll
        for (int kc = 0; kc < 16; ++kc) {
            const float4 q = wcp[(2 * kc + h) * NA + selidx];
            a_re[kc].x -= sre * q.x + sim * q.z;
            a_im[kc].x -= sim * q.x - sre * q.z;
            a_re[kc].y -= sre * q.y + sim * q.w;
            a_im[kc].y -= sim * q.y - sre * q.w;
        }
    }

    // Emit residual and x_hat = x - residual (interleaved complex64).
#pragma unroll
    for (int kc = 0; kc < 16; ++kc) {
        const int k0 = 4 * kc + 2 * h;
        const long off = (b * NN + k0) * 2;
        const float4 r = {a_re[kc].x, a_im[kc].x, a_re[kc].y, a_im[kc].y};
        *(float4*)(out_res + off) = r;
        const float4 xv = *(const float4*)(x + off);
        const float4 xh = {xv.x - r.x, xv.y - r.y, xv.z - r.z, xv.w - r.w};
        *(float4*)(out_xhat + off) = xh;
    }
}

// ---------------------------------------------------------------------------
extern "C" void kernel_launch(void* const* d_in, const int* in_sizes, int n_in,
                              void* d_out, int out_size, void* d_ws, size_t ws_size,
                              hipStream_t stream) {
    (void)in_sizes; (void)n_in; (void)out_size; (void)ws_size;

    const float* x      = (const float*)d_in[0];   // complex64 [8192,64] interleaved
    const float* antpos = (const float*)d_in[1];   // f32 [64,3]
    const float* doa    = (const float*)d_in[2];   // f32 [1024,3]
    const float* gvec   = (const float*)d_in[3];   // f32 [64]
    const int*   kptr   = (const int*)d_in[4];     // scalar k

    float* wcp = (float*)d_ws;                     // packed dict, 512 KB

    float* out_res  = (float*)d_out;               // residual, c64 interleaved
    float* out_xhat = out_res + (long)NB * NN * 2; // x_hat

    // Build packed conj(W) dictionary: 64*1024 elements.
    build_wc_kernel<<<(NN * NA) / 256, 256, 0, stream>>>(antpos, doa, gvec, wcp);

    // Fused k-iteration matching pursuit: 128 blocks x 4 waves x 16 rows = 8192.
    mp_fused_kernel<<<NB / 64, 128, 0, stream>>>(x, (const float4*)wcp, kptr,
                                                 out_res, out_xhat);
}